// MSE_plus_physics_loss_54631984005299
// MI455X (gfx1250) — compile-verified
//
#include <hip/hip_runtime.h>
#include <hip/hip_bf16.h>
#include <stdint.h>

#define N_BUSES 2000
#define NBATCH  32
#define NEDGE   6000
#define BN      (NBATCH * N_BUSES)

typedef __attribute__((ext_vector_type(2))) float v2f;
typedef __attribute__((ext_vector_type(8))) float v8f;

// workspace layout (floats)
#define ACC_OFF 0     // [0..4]: sum d1re, d1im, d2, d3, mse ; [5..7] pad
#define SRE_OFF 8     // 32: per-batch mean(S.re)
#define SIM_OFF 40    // 32: per-batch mean(S.im)
#define VMM_OFF 72    // 32: per-batch mean(Vm)
#define YV_OFF  128   // 2*BN floats: YV interleaved (re,im)

// ---------- wave32 sum via V_WMMA_F32_16X16X4_F32 ----------
// A = {v, 0} per lane (covers all 64 A-slots across the wave), B = ones.
// D[m,n] = rowsum_A(m) for every column n. Lane L sums its 8 D VGPRs
// (rows 0-7 for lanes<16, rows 8-15 for lanes>=16 per documented C/D layout),
// then xor-16 shuffle completes the 32-lane total on every lane.
__device__ __forceinline__ float wave_sum_wmma(float v) {
  v2f a = {v, 0.0f};
  v2f b = {1.0f, 1.0f};
  v8f c = {};
  v8f d = __builtin_amdgcn_wmma_f32_16x16x4_f32(false, a, false, b,
                                                (short)0, c, false, false);
  float s = d[0] + d[1] + d[2] + d[3] + d[4] + d[5] + d[6] + d[7];
  s += __shfl_xor(s, 16, 32);
  return s;
}

__device__ __forceinline__ int wrapN(int v) {
  int m = v % N_BUSES;
  return (m < 0) ? m + N_BUSES : m;
}

// ---------- zero the scalar accumulators ----------
__global__ void init_kernel(float* ws) {
  if (threadIdx.x < 8) ws[ACC_OFF + threadIdx.x] = 0.0f;
}

// ---------- per-batch means of S.re, S.im, Vm ----------
__global__ void means_kernel(const float* __restrict__ x, float* __restrict__ ws) {
  __shared__ float r0[256], r1[256], r2[256];
  const int b = blockIdx.x, tid = threadIdx.x;
  float a0 = 0.0f, a1 = 0.0f, a2 = 0.0f;
  for (int n = tid; n < N_BUSES; n += 256) {
    const float* xr = x + (size_t)(b * N_BUSES + n) * 6;
    a0 += xr[0]; a1 += xr[1]; a2 += xr[2];
  }
  r0[tid] = a0; r1[tid] = a1; r2[tid] = a2;
  __syncthreads();
  for (int s = 128; s > 0; s >>= 1) {
    if (tid < s) { r0[tid] += r0[tid+s]; r1[tid] += r1[tid+s]; r2[tid] += r2[tid+s]; }
    __syncthreads();
  }
  if (tid == 0) {
    const float inv = 1.0f / (float)N_BUSES;
    ws[SRE_OFF + b] = r0[0] * inv;
    ws[SIM_OFF + b] = r1[0] * inv;
    ws[VMM_OFF + b] = r2[0] * inv;
  }
}

// ---------- sparse Y*V per batch: async-stage V to LDS, LDS scatter-add ----------
__global__ void scatter_kernel(const float* __restrict__ edge_attr,
                               const int*   __restrict__ edge_index,
                               const float* __restrict__ outputs,
                               float* __restrict__ ws) {
  __shared__ float ldsV[2 * N_BUSES];   // staged V (re,im) for this batch
  __shared__ float ldsA[2 * N_BUSES];   // YV accumulators
  const int b   = blockIdx.x;
  const int tid = threadIdx.x;
  const int nt  = blockDim.x;

  for (int k = tid; k < 2 * N_BUSES; k += nt) ldsA[k] = 0.0f;

  // Async DMA: outputs row (re,im) is 8 contiguous bytes -> one b64 per bus.
  for (int n = tid; n < N_BUSES; n += nt) {
    unsigned long long gaddr =
        (unsigned long long)(uintptr_t)(outputs + (size_t)(b * N_BUSES + n) * 2);
    unsigned loff = (unsigned)(uintptr_t)(&ldsV[2 * n]);
    asm volatile("global_load_async_to_lds_b64 %0, %1, off"
                 :: "v"(loff), "v"(gaddr)
                 : "memory");
  }
  asm volatile("s_wait_asynccnt 0" ::: "memory");
  __syncthreads();

  // Scatter: YV[r] += (g + j*bb) * V[c]   (sum for duplicate (r,c); harmless here)
  for (int e = tid; e < NEDGE; e += nt) {
    const int idx = b * NEDGE + e;
    const int r = wrapN(edge_index[idx]);
    const int c = wrapN(edge_index[NBATCH * NEDGE + idx]);
    const float g  = edge_attr[2 * idx + 0];
    const float bb = edge_attr[2 * idx + 1];
    const float vr = ldsV[2 * c + 0];
    const float vi = ldsV[2 * c + 1];
    atomicAdd(&ldsA[2 * r + 0], g * vr - bb * vi);
    atomicAdd(&ldsA[2 * r + 1], g * vi + bb * vr);
  }
  __syncthreads();

  float* yv = ws + YV_OFF;
  for (int n = tid; n < N_BUSES; n += nt) {
    yv[2 * (b * N_BUSES + n) + 0] = ldsA[2 * n + 0];
    yv[2 * (b * N_BUSES + n) + 1] = ldsA[2 * n + 1];
  }
}

// ---------- per-bus terms + WMMA wave reduction -> 5 global accumulators ----------
__global__ void bus_kernel(const float* __restrict__ x,
                           const float* __restrict__ outputs,
                           const float* __restrict__ labels,
                           float* __restrict__ ws) {
  const int i = blockIdx.x * blockDim.x + threadIdx.x;   // grid sized exactly BN
  const int b = i / N_BUSES;

  const float* xr = x + (size_t)i * 6;
  const float sre = xr[0], sim = xr[1], vm = xr[2];
  const float bt0 = xr[3], bt1 = xr[4], bt2 = xr[5];
  const float vr = outputs[2 * i + 0], vi = outputs[2 * i + 1];
  const float* yv = ws + YV_OFF;
  const float yr = yv[2 * i + 0], yi = yv[2 * i + 1];

  // Spred = V * conj(YV)
  const float sp_re = vr * yr + vi * yi;
  const float sp_im = vi * yr - vr * yi;

  // |1/S| (fallback to |1/S_mean| when S == 0)
  float sinv;
  if (sre != 0.0f || sim != 0.0f) {
    sinv = 1.0f / sqrtf(sre * sre + sim * sim);
  } else {
    const float mre = ws[SRE_OFF + b], mim = ws[SIM_OFF + b];
    sinv = 1.0f / sqrtf(mre * mre + mim * mim);
  }

  const float dre = sp_re - sre, dim_ = sp_im - sim;
  const float r1re = (dre * dre - dim_ * dim_) * bt0;   // complex square, real
  const float r1im = 2.0f * dre * dim_ * bt0;           // complex square, imag
  const float r2   = dre * dre * bt1;                   // real part diff squared
  const float p0 = sinv * (r1re + r2);                  // D1.re
  const float p1 = sinv * r1im;                         // D1.im

  const float vminv = (vm != 0.0f) ? (1.0f / vm) : (1.0f / ws[VMM_OFF + b]);
  const float vmag  = sqrtf(vr * vr + vi * vi);
  const float p2 = fabsf(vmag * (bt1 + bt2) - vm) * vminv;  // D2
  const float p3 = fabsf(vi * bt2) * vminv;                 // D3

  const float e0 = vr - labels[2 * i + 0];
  const float e1 = vi - labels[2 * i + 1];
  const float p4 = e0 * e0 + e1 * e1;                       // mse partial

  // wave32 reductions on the matrix pipe (EXEC all ones: exact grid sizing,
  // called unconditionally)
  const float s0 = wave_sum_wmma(p0);
  const float s1 = wave_sum_wmma(p1);
  const float s2 = wave_sum_wmma(p2);
  const float s3 = wave_sum_wmma(p3);
  const float s4 = wave_sum_wmma(p4);

  __shared__ float red[8][8];
  const int wave = threadIdx.x >> 5;
  const int lane = threadIdx.x & 31;
  if (lane == 0) {
    red[wave][0] = s0; red[wave][1] = s1; red[wave][2] = s2;
    red[wave][3] = s3; red[wave][4] = s4;
  }
  __syncthreads();
  if (threadIdx.x < 5) {
    float t = 0.0f;
#pragma unroll
    for (int w = 0; w < 8; ++w) t += red[w][threadIdx.x];
    atomicAdd(&ws[ACC_OFF + threadIdx.x], t);
  }
}

// ---------- finalize: 5 scalars (complex parts expanded if out_size allows) ----------
__global__ void finalize_kernel(const float* __restrict__ ws,
                                float* __restrict__ out, int out_size) {
  if (threadIdx.x != 0 || blockIdx.x != 0) return;
  const float inv = 1.0f / (float)BN;
  const float d1re = ws[0] * inv, d1im = ws[1] * inv;
  const float d2 = ws[2] * inv, d3 = ws[3] * inv;
  const float mse = ws[4] * (0.5f * inv);
  const float pre = d1re + d2 + d3;   // W1=W2=W3=1
  const float pim = d1im;
  const float lre = mse + 0.1f * pre; // LMBDA=0.1
  const float lim = 0.1f * pim;
  if (out_size >= 8) {
    out[0] = lre;  out[1] = lim;
    out[2] = pre;  out[3] = pim;
    out[4] = d1re; out[5] = d1im;
    out[6] = d2;   out[7] = d3;
  } else {
    const float vals[5] = {lre, pre, d1re, d2, d3};
    for (int k = 0; k < out_size && k < 5; ++k) out[k] = vals[k];
  }
}

extern "C" void kernel_launch(void* const* d_in, const int* in_sizes, int n_in,
                              void* d_out, int out_size, void* d_ws, size_t ws_size,
                              hipStream_t stream) {
  (void)in_sizes; (void)n_in; (void)ws_size;
  const float* x         = (const float*)d_in[0];
  const float* edge_attr = (const float*)d_in[1];
  const int*   edge_index= (const int*)d_in[2];
  const float* outputs   = (const float*)d_in[3];
  const float* labels    = (const float*)d_in[4];
  float* ws  = (float*)d_ws;
  float* out = (float*)d_out;

  hipLaunchKernelGGL(init_kernel,    dim3(1),        dim3(32),  0, stream, ws);
  hipLaunchKernelGGL(means_kernel,   dim3(NBATCH),   dim3(256), 0, stream, x, ws);
  hipLaunchKernelGGL(scatter_kernel, dim3(NBATCH),   dim3(512), 0, stream,
                     edge_attr, edge_index, outputs, ws);
  hipLaunchKernelGGL(bus_kernel,     dim3(BN / 256), dim3(256), 0, stream,
                     x, outputs, labels, ws);
  hipLaunchKernelGGL(finalize_kernel, dim3(1), dim3(1), 0, stream, out ? ws : ws, out, out_size);
}